// SumLayer_63823214019292
// MI455X (gfx1250) — compile-verified
//
#include <hip/hip_runtime.h>
#include <hip/hip_bf16.h>
#include <cstdint>

// SumLayer forward: per node, weighted logsumexp over 32 gathered child rows.
//
// MI455X design:
//  - 512 MB of 512B row-gathers; element_mars (128 MB) is L2-resident (192 MB L2)
//    -> gather-latency / L2-bandwidth bound. FLOPs (~134M exp) are negligible.
//  - CDNA5 async global->LDS DMA (ASYNCcnt): 32 gathers in flight per wave with
//    zero destination VGPRs; one b64 async (8B/lane, 256B/row) per child row.
//  - blockDim = 64 (2 waves), ONE node per block (node = blockIdx.x uniform).
//  - Index distribution avoids the SMEM/kmcnt path entirely: lane c of the wave
//    holds cid/pid of child c (one coalesced vector load each), and per-child
//    values are broadcast to SGPRs with v_readlane (wave32) — no per-child
//    constant-cache round-trips on the issue path.
//  - params gathered as one per-lane vector load that overlaps the async DMA;
//    weights broadcast by readlane in pass 2.
//  - Each lane reads back only LDS its own wave's asyncs produced, so
//    s_wait_asynccnt 0 is the only synchronization (no barriers).
//  - 16 KB LDS/block -> 20 blocks/WGP -> occupancy wave-slot-limited.

#define MAX_CHS 32
#define BATCH   128
#define EPS     1e-10f

__launch_bounds__(64)
__global__ void sum_layer_fwd_kernel(const float* __restrict__ element_mars,
                                     const float* __restrict__ params,
                                     const unsigned* __restrict__ cids,
                                     const unsigned* __restrict__ pids,
                                     float* __restrict__ out) {
    // [wave][child][lane] float2 staging: 2 * 32 * 32 * 8B = 16 KB
    __shared__ float2 smem[2 * MAX_CHS * 32];

    const int lane = threadIdx.x & 31;
    const int half = threadIdx.x >> 5;       // which 64-column half of the batch
    const int node = blockIdx.x;
    const int col  = half * 64 + lane * 2;   // first of this lane's 2 columns

    float2* __restrict__ wbuf = smem + half * (MAX_CHS * 32);  // wave-private 8 KB

    // ---- Wave-register index file: lane c holds child c's ids ----
    const unsigned cidv = cids[node * MAX_CHS + lane];   // 1 coalesced load
    const unsigned pidv = pids[node * MAX_CHS + lane];   // 1 coalesced load
    const float    wv   = params[pidv];                  // 1 gathered load
                                                         // (overlaps async DMA)

    // ---- Issue 32 async row-gathers: GLOBAL_LOAD_ASYNC_TO_LDS_B64 (GVS) ----
    const unsigned voff = (unsigned)(col * 4);   // loop-invariant VGPR byte offset
    #pragma unroll
    for (int c = 0; c < MAX_CHS; ++c) {
        // Broadcast child id into an SGPR: single VALU op, no memory latency.
        unsigned cid_c = (unsigned)__builtin_amdgcn_readlane((int)cidv, c);
        const float* srow = element_mars + (size_t)cid_c * BATCH;    // SGPR pair
        unsigned lds = (unsigned)(uintptr_t)(&wbuf[c * 32 + lane]);  // LDS offset
        asm volatile("global_load_async_to_lds_b64 %0, %1, %2"
                     :: "v"(lds), "v"(voff), "s"(srow));
    }

    // DMA completion barrier: pins all LDS reads below this point.
    asm volatile("s_wait_asynccnt 0" ::: "memory");

    // ---- Pass 1: per-column max over children ----
    float2 m = make_float2(-__builtin_inff(), -__builtin_inff());
    #pragma unroll
    for (int c = 0; c < MAX_CHS; ++c) {
        float2 x = wbuf[c * 32 + lane];
        m.x = fmaxf(m.x, x.x);
        m.y = fmaxf(m.y, x.y);
    }

    // ---- Pass 2: weighted exp-sum (weights broadcast via readlane) ----
    float2 s = make_float2(0.0f, 0.0f);
    #pragma unroll
    for (int c = 0; c < MAX_CHS; ++c) {
        float w_c = __uint_as_float(
            (unsigned)__builtin_amdgcn_readlane((int)__float_as_uint(wv), c));
        float2 x = wbuf[c * 32 + lane];
        s.x = __builtin_fmaf(__expf(x.x - m.x), w_c, s.x);
        s.y = __builtin_fmaf(__expf(x.y - m.y), w_c, s.y);
    }

    float2 r;
    r.x = __logf(fmaxf(s.x, EPS)) + m.x;
    r.y = __logf(fmaxf(s.y, EPS)) + m.y;
    *(float2*)(out + (size_t)node * BATCH + col) = r;
}

extern "C" void kernel_launch(void* const* d_in, const int* in_sizes, int n_in,
                              void* d_out, int out_size, void* d_ws, size_t ws_size,
                              hipStream_t stream) {
    const float*    element_mars = (const float*)d_in[0];
    const float*    params       = (const float*)d_in[1];
    const unsigned* cids         = (const unsigned*)d_in[2];
    const unsigned* pids         = (const unsigned*)d_in[3];
    float*          out          = (float*)d_out;

    const int num_nodes = in_sizes[2] / MAX_CHS;   // 32768

    sum_layer_fwd_kernel<<<num_nodes, 64, 0, stream>>>(
        element_mars, params, cids, pids, out);
}